// GaussianInpainter_28226525070080
// MI455X (gfx1250) — compile-verified
//
#include <hip/hip_runtime.h>

// GaussianInpainter for MI455X (gfx1250).
// Separable Gaussian (K=7, sigma=5) applied twice along each of D,H,W.
// Per-axis truncated convolutions commute across axes, so the 6 reference
// passes compose exactly into 3 passes of the squared operator T^2 (13-tap
// band, boundary-corrected coefficients). Each wave computes a 16x32 output
// tile as two banded GEMMs: D[16x16] = A[16x28] * B[28x16], via 7 chained
// V_WMMA_F32_16X16X4_F32 per tile (14 WMMA / wave, full f32 precision).
//
// d_ws layout: bufA (4*192^3 f32) | bufB (4*192^3 f32) | sums (4 f32)
// Traffic ~1.3 GB -> ~55 us floor @ 23.3 TB/s; ping/pong mostly L2-resident.

#define NDIM   192
#define VCUBE  (192 * 192 * 192)          // 7,077,888
#define LINES_PER_BC (192 * 192)          // 36,864

typedef __attribute__((ext_vector_type(2))) float v2f;
typedef __attribute__((ext_vector_type(4))) float v4f;
typedef __attribute__((ext_vector_type(8))) float v8f;

// 7-tap Gaussian, sigma=5, normalized (matches reference _gaussian_1d).
__device__ __constant__ float W7[7] = {
    0.12895603f, 0.14251846f, 0.15133131f, 0.15438840f,
    0.15133131f, 0.14251846f, 0.12895603f};

// Interior composed kernel: W13 = W7 (*) W7 (exact for outputs >=6 cells
// from either border).
__device__ __constant__ float W13[13] = {
    0.0166297f, 0.0367572f, 0.0593417f, 0.0829536f, 0.1059378f, 0.1266198f,
    0.1435204f,
    0.1266198f, 0.1059378f, 0.0829536f, 0.0593417f, 0.0367572f, 0.0166297f};

// Boundary-corrected coefficient of T^2 at output index i, tap offset t:
// C(i,t) = sum_s W7[s+3]*W7[t-s+3] with intermediate index i+s clipped to
// [0,191] (zero-padding truncation of the first 7-tap pass).
__device__ __forceinline__ float wcoef(int i, int t) {
    int lo = -3;
    if (t - 3 > lo) lo = t - 3;
    if (-i > lo)    lo = -i;
    int hi = 3;
    if (t + 3 < hi)        hi = t + 3;
    if (NDIM - 1 - i < hi) hi = NDIM - 1 - i;
    float s = 0.0f;
    for (int q = lo; q <= hi; ++q)
        s += W7[q + 3] * W7[t - q + 3];
    return s;
}

__global__ __launch_bounds__(256) void zero_sums(float* sums) {
    if (threadIdx.x < 4) sums[threadIdx.x] = 0.0f;
}

// Per-(b,c) spatial sums of the UNFILTERED u (epsilon source).
__global__ __launch_bounds__(256) void sum_kernel(const float* __restrict__ x,
                                                  const float* __restrict__ m,
                                                  float* __restrict__ sums) {
    __shared__ float red[256];
    int idx = blockIdx.x * 256 + threadIdx.x;   // over 4*VCUBE (VCUBE%256==0)
    int bc  = idx / VCUBE;
    int v   = idx - bc * VCUBE;
    int b   = bc >> 1;
    float val = (bc & 1) ? m[b * VCUBE + v] : x[b * VCUBE + v];
    red[threadIdx.x] = val;
    __syncthreads();
    for (int s = 128; s > 0; s >>= 1) {
        if (threadIdx.x < s) red[threadIdx.x] += red[threadIdx.x + s];
        __syncthreads();
    }
    if (threadIdx.x == 0) atomicAdd(&sums[bc], red[0]);
}

// One pass of T^2 along one axis; compile-time strides so interior loads use
// immediate offsets. One wave handles 16 outputs x 32 lines (two 16x16 WMMA
// tiles sharing the A band). srcU==nullptr => first pass reads x/mask direct.
template <int AS, int OS, int IS>
__global__ __launch_bounds__(256) void conv_pass(
    const float* __restrict__ srcX, const float* __restrict__ srcM,
    const float* __restrict__ srcU, float* __restrict__ dst) {
    const int lane  = threadIdx.x & 31;
    const int wave  = blockIdx.x * 8 + (threadIdx.x >> 5);  // 0..55295
    const int tileA = wave % 12;
    const int tileL = wave / 12;                            // 0..4607
    const int m0    = tileA << 4;

    const int l0     = tileL << 5;          // 32 consecutive lines
    const int bc     = l0 / LINES_PER_BC;   // uniform across tile
    const int r      = l0 - bc * LINES_PER_BC;
    const int outer  = r / NDIM;            // uniform across tile
    const int inner0 = r - outer * NDIM;

    const float* src;
    int srcOff;
    if (srcU) { src = srcU; srcOff = bc * VCUBE; }
    else      { src = (bc & 1) ? srcM : srcX; srcOff = (bc >> 1) * VCUBE; }

    const int n     = lane & 15;            // line / M index within tile
    const int kHalf = (lane >> 4) << 1;     // 0 | 2  (f32 WMMA A/B layout)
    const int iAbs  = m0 + n;

    const int lineOff0 = outer * OS + (inner0 + n) * IS;
    const int lineOff1 = lineOff0 + 16 * IS;

    const bool interior = (tileA >= 1) && (tileA <= 10);   // wave-uniform

    // ---- A fragments: banded composed-Gaussian weights ----
    // Layout: lanes 0-15 -> M=lane, K=kHalf+{0,1} in VGPR0/1; lanes 16-31
    // -> M=lane-16, K=kHalf+{0,1} with kHalf=2.
    v2f aFrag[7];
    if (interior) {
#pragma unroll
        for (int j = 0; j < 7; ++j) {
            int t0 = 4 * j + kHalf - n - 6;
            int t1 = t0 + 1;
            float a0 = (t0 >= -6 && t0 <= 6) ? W13[t0 + 6] : 0.0f;
            float a1 = (t1 >= -6 && t1 <= 6) ? W13[t1 + 6] : 0.0f;
            aFrag[j] = (v2f){a0, a1};
        }
    } else {
#pragma unroll
        for (int j = 0; j < 7; ++j) {
            int t0 = 4 * j + kHalf - n - 6;
            int t1 = t0 + 1;
            float a0 = (t0 >= -6 && t0 <= 6) ? wcoef(iAbs, t0) : 0.0f;
            float a1 = (t1 >= -6 && t1 <= 6) ? wcoef(iAbs, t1) : 0.0f;
            aFrag[j] = (v2f){a0, a1};
        }
    }

    // ---- B fragments: input windows for both line tiles (hoisted) ----
    v2f bf0[7], bf1[7];
    if (interior) {
        // Whole window m0-6..m0+21 in range: unconditional loads, immediate
        // offsets (strides are compile-time).
        const float* s0 = src + srcOff + lineOff0 + (m0 - 6 + kHalf) * AS;
        const float* s1 = src + srcOff + lineOff1 + (m0 - 6 + kHalf) * AS;
        if constexpr (AS == 1) {
#pragma unroll
            for (int j = 0; j < 7; ++j) {       // contiguous pair: b64 loads
                bf0[j] = *(const v2f*)(s0 + 4 * j);
                bf1[j] = *(const v2f*)(s1 + 4 * j);
            }
        } else {
#pragma unroll
            for (int j = 0; j < 7; ++j) {
                bf0[j] = (v2f){s0[(4 * j) * AS], s0[(4 * j + 1) * AS]};
                bf1[j] = (v2f){s1[(4 * j) * AS], s1[(4 * j + 1) * AS]};
            }
        }
    } else {
        // Edge tiles: clamp address, select-to-zero (no exec divergence).
        const float* s0 = src + srcOff + lineOff0;
        const float* s1 = src + srcOff + lineOff1;
#pragma unroll
        for (int j = 0; j < 7; ++j) {
            int p0 = m0 - 6 + kHalf + 4 * j;
            int p1 = p0 + 1;
            int q0 = min(max(p0, 0), NDIM - 1);
            int q1 = min(max(p1, 0), NDIM - 1);
            float v00 = s0[q0 * AS], v01 = s0[q1 * AS];
            float v10 = s1[q0 * AS], v11 = s1[q1 * AS];
            bf0[j] = (v2f){p0 == q0 ? v00 : 0.0f, p1 == q1 ? v01 : 0.0f};
            bf1[j] = (v2f){p0 == q0 ? v10 : 0.0f, p1 == q1 ? v11 : 0.0f};
        }
    }

    // ---- 14 chained WMMAs (two independent accumulator chains) ----
    v8f acc0 = {}, acc1 = {};
#pragma unroll
    for (int j = 0; j < 7; ++j) {
        acc0 = __builtin_amdgcn_wmma_f32_16x16x4_f32(
            false, aFrag[j], false, bf0[j], (short)0, acc0, false, false);
        acc1 = __builtin_amdgcn_wmma_f32_16x16x4_f32(
            false, aFrag[j], false, bf1[j], (short)0, acc1, false, false);
    }

    // ---- store: C/D layout VGPR rr -> M = rr + 8*(lane>=16), N = lane&15 ----
    const int mBase = (lane >> 4) << 3;
    const int d0 = bc * VCUBE + lineOff0;
    const int d1 = bc * VCUBE + lineOff1;
    if constexpr (AS == 1) {
        v4f* o0 = (v4f*)(dst + d0 + m0 + mBase);   // 32B-aligned
        v4f* o1 = (v4f*)(dst + d1 + m0 + mBase);
        o0[0] = (v4f){acc0[0], acc0[1], acc0[2], acc0[3]};
        o0[1] = (v4f){acc0[4], acc0[5], acc0[6], acc0[7]};
        o1[0] = (v4f){acc1[0], acc1[1], acc1[2], acc1[3]};
        o1[1] = (v4f){acc1[4], acc1[5], acc1[6], acc1[7]};
    } else {
#pragma unroll
        for (int rr = 0; rr < 8; ++rr) {
            dst[d0 + (m0 + mBase + rr) * AS] = acc0[rr];
            dst[d1 + (m0 + mBase + rr) * AS] = acc1[rr];
        }
    }
}

__global__ __launch_bounds__(256) void finalize_kernel(
    const float* __restrict__ u, const float* __restrict__ sums,
    float* __restrict__ out) {
    int idx = blockIdx.x * 256 + threadIdx.x;   // over 2*VCUBE
    int b   = idx / VCUBE;
    int v   = idx - b * VCUBE;
    float e0 = sums[b * 2 + 0] * 1e-8f;
    float e1 = sums[b * 2 + 1] * 1e-8f;
    float fx = u[(b * 2 + 0) * VCUBE + v] + e0;
    float fm = u[(b * 2 + 1) * VCUBE + v] + e1;
    out[idx] = fx / fm;
}

extern "C" void kernel_launch(void* const* d_in, const int* in_sizes, int n_in,
                              void* d_out, int out_size, void* d_ws, size_t ws_size,
                              hipStream_t stream) {
    const float* x    = (const float*)d_in[0];
    const float* mask = (const float*)d_in[1];
    float*       out  = (float*)d_out;

    float* bufA = (float*)d_ws;
    float* bufB = bufA + (size_t)4 * VCUBE;
    float* sums = bufB + (size_t)4 * VCUBE;

    zero_sums<<<1, 256, 0, stream>>>(sums);
    sum_kernel<<<(4 * VCUBE) / 256, 256, 0, stream>>>(x, mask, sums);

    // 12 axis tiles * (4*36864/32) line tiles = 55296 waves; 8 waves/block.
    const int convBlocks = 55296 / 8;   // 6912
    // Pass 1: axis D (stride 192^2); lines over (h,w) -> inner stride 1.
    conv_pass<NDIM * NDIM, NDIM, 1>
        <<<convBlocks, 256, 0, stream>>>(x, mask, nullptr, bufA);
    // Pass 2: axis H (stride 192); lines over (d,w) -> inner stride 1.
    conv_pass<NDIM, NDIM * NDIM, 1>
        <<<convBlocks, 256, 0, stream>>>(nullptr, nullptr, bufA, bufB);
    // Pass 3: axis W (stride 1); lines over (d,h) -> inner stride 192.
    conv_pass<1, NDIM * NDIM, NDIM>
        <<<convBlocks, 256, 0, stream>>>(nullptr, nullptr, bufB, bufA);

    finalize_kernel<<<(2 * VCUBE) / 256, 256, 0, stream>>>(bufA, sums, out);
}